// TopDownModel_69836168233106
// MI455X (gfx1250) — compile-verified
//
#include <hip/hip_runtime.h>
#include <hip/hip_bf16.h>
#include <math.h>

// ---------------------------------------------------------------------------
// TopDown attention caption step, MI455X (gfx1250, wave32).
// Memory-bound (~200MB weight streaming); GEMVs are coalesced float4 streams,
// the one real GEMM (img @ Wf^T, 36x512x2048) uses V_WMMA_F32_16X16X4_F32.
// ---------------------------------------------------------------------------

#define WAVE 32

typedef __attribute__((ext_vector_type(2))) float v2f;
typedef __attribute__((ext_vector_type(8))) float v8f;

__device__ __forceinline__ float sigmoidf(float x) {
    return 1.0f / (1.0f + __expf(-x));
}

// ----------------------------- concat x_att --------------------------------
// x_att[0:1000) = language_lstm_h ; x_att[1000:3048) = pooled_img_features
__global__ void concat_xatt_kernel(const float* __restrict__ h_lang_prev,
                                   const float* __restrict__ pooled,
                                   float* __restrict__ x_att) {
    int i = blockIdx.x * blockDim.x + threadIdx.x;
    if (i < 1000)       x_att[i] = h_lang_prev[i];
    else if (i < 3048)  x_att[i] = pooled[i - 1000];
}

// ----------------------------- generic GEMV --------------------------------
// y[row] = dot(W[row, 0:K), x) + b0[row] + b1[row]   (biases nullable)
// One 256-thread block per row; float4 streaming loads; wave32 shuffle reduce.
__global__ void gemv_kernel(const float* __restrict__ W,
                            const float* __restrict__ x,
                            const float* __restrict__ b0,
                            const float* __restrict__ b1,
                            float* __restrict__ y, int K) {
    const int row = blockIdx.x;
    const float4* __restrict__ Wr = reinterpret_cast<const float4*>(W + (size_t)row * K);
    const float4* __restrict__ xv = reinterpret_cast<const float4*>(x);
    const int K4 = K >> 2;

    float acc = 0.0f;
    for (int i = threadIdx.x; i < K4; i += blockDim.x) {
        float4 w = Wr[i];
        float4 v = xv[i];
        acc = fmaf(w.x, v.x, acc);
        acc = fmaf(w.y, v.y, acc);
        acc = fmaf(w.z, v.z, acc);
        acc = fmaf(w.w, v.w, acc);
    }
    // wave32 reduce
    #pragma unroll
    for (int off = WAVE / 2; off > 0; off >>= 1)
        acc += __shfl_down(acc, off, WAVE);

    __shared__ float red[8]; // 256 / 32
    const int lane = threadIdx.x & (WAVE - 1);
    const int wid  = threadIdx.x >> 5;
    if (lane == 0) red[wid] = acc;
    __syncthreads();
    if (threadIdx.x == 0) {
        float s = 0.0f;
        const int nw = blockDim.x >> 5;
        for (int i = 0; i < nw; ++i) s += red[i];
        if (b0) s += b0[row];
        if (b1) s += b1[row];
        y[row] = s;
    }
}

// ------------------------ LSTM activation (zero state) ---------------------
// gates = [i | f | g | o] each length H; c0=0 so c = sig(i)*tanh(g),
// h = sig(o)*tanh(c)
__global__ void lstm_act_kernel(const float* __restrict__ gates,
                                float* __restrict__ h, int H) {
    int j = blockIdx.x * blockDim.x + threadIdx.x;
    if (j < H) {
        float ig = gates[j];
        float g  = gates[2 * H + j];
        float o  = gates[3 * H + j];
        float c  = sigmoidf(ig) * tanhf(g);
        h[j]     = sigmoidf(o) * tanhf(c);
    }
}

// ------------------- wf GEMM via V_WMMA_F32_16X16X4_F32 --------------------
// wf(36x512) = img(36x2048) @ Wf^T(2048x512).  One wave per 16x16 D tile.
// f32 A 16x4 layout: lanes 0-15 -> {K=0 (v0), K=1 (v1)}, lanes 16-31 -> {K=2, K=3}
// f32 B 4x16 layout: mirrored (N = lane&15, K half by lane>=16)
// f32 D 16x16: lane<16 -> N=lane, M=v ; lane>=16 -> N=lane-16, M=v+8
__global__ __launch_bounds__(32)
void wf_wmma_kernel(const float* __restrict__ img,  // (36, 2048)
                    const float* __restrict__ Wf,   // (512, 2048)
                    float* __restrict__ wf) {       // (36, 512)
    constexpr int R = 36, D = 2048, A = 512;
    const int ntile = blockIdx.x;       // 0..31
    const int mtile = blockIdx.y;       // 0..2
    const int lane  = threadIdx.x;      // 0..31
    const int half  = lane >> 4;        // K-half select
    const int l     = lane & 15;

    const int m = mtile * 16 + l;       // A row this lane streams
    const int n = ntile * 16 + l;       // B column (= Wf row) this lane streams
    const bool mvalid = (m < R);
    const float msk = mvalid ? 1.0f : 0.0f;
    // clamp pointer so masked lanes still read in-bounds (EXEC stays all-1s)
    const float* __restrict__ arow = img + (size_t)(mvalid ? m : (R - 1)) * D;
    const float* __restrict__ brow = Wf + (size_t)n * D;   // B[k][n] = Wf[n][k]

    v8f acc = {};
    for (int k0 = 0; k0 < D; k0 += 4) {
        const int ka = k0 + half * 2;
        v2f afrag, bfrag;
        float2 av = *reinterpret_cast<const float2*>(arow + ka);
        float2 bv = *reinterpret_cast<const float2*>(brow + ka);
        afrag[0] = msk * av.x;
        afrag[1] = msk * av.y;
        bfrag[0] = bv.x;
        bfrag[1] = bv.y;
        acc = __builtin_amdgcn_wmma_f32_16x16x4_f32(
            /*neg_a=*/false, afrag, /*neg_b=*/false, bfrag,
            /*c_mod=*/(short)0, acc, /*reuse_a=*/false, /*reuse_b=*/false);
    }

    const int ncol  = ntile * 16 + l;
    const int mbase = mtile * 16 + half * 8;
    #pragma unroll
    for (int v = 0; v < 8; ++v) {
        const int mm = mbase + v;
        if (mm < R) wf[(size_t)mm * A + ncol] = acc[v];
    }
}

// ----------------------- attention score per region ------------------------
// aw[r] = sum_a Wa[a] * tanh(wf[r][a] + wh[a])       (A = 512)
__global__ void att_score_kernel(const float* __restrict__ wf,
                                 const float* __restrict__ wh,
                                 const float* __restrict__ Wa,
                                 float* __restrict__ aw) {
    const int r = blockIdx.x;   // 0..35
    float acc = 0.0f;
    for (int a = threadIdx.x; a < 512; a += blockDim.x)
        acc += Wa[a] * tanhf(wf[r * 512 + a] + wh[a]);

    #pragma unroll
    for (int off = WAVE / 2; off > 0; off >>= 1)
        acc += __shfl_down(acc, off, WAVE);

    __shared__ float red[8];
    const int lane = threadIdx.x & (WAVE - 1);
    const int wid  = threadIdx.x >> 5;
    if (lane == 0) red[wid] = acc;
    __syncthreads();
    if (threadIdx.x == 0) {
        float s = 0.0f;
        const int nw = blockDim.x >> 5;
        for (int i = 0; i < nw; ++i) s += red[i];
        aw[r] = s;
    }
}

// --------------------- attended features + concat x_lang -------------------
// x_lang[0:2048)  = sum_r aw[r] * img[r][d]
// x_lang[2048:3048) = h_att
__global__ void attend_kernel(const float* __restrict__ img,
                              const float* __restrict__ aw,
                              const float* __restrict__ h_att,
                              float* __restrict__ x_lang) {
    int d = blockIdx.x * blockDim.x + threadIdx.x;
    if (d < 2048) {
        float s = 0.0f;
        #pragma unroll 4
        for (int r = 0; r < 36; ++r)
            s = fmaf(aw[r], img[r * 2048 + d], s);
        x_lang[d] = s;
    } else if (d < 3048) {
        x_lang[d] = h_att[d - 2048];
    }
}

// ------------------------------- softmax -----------------------------------
// Single 1024-thread block over N=10000 logits.
__global__ __launch_bounds__(1024)
void softmax_kernel(const float* __restrict__ logits,
                    float* __restrict__ out, int N) {
    __shared__ float red[32];
    __shared__ float bcast;
    const int tid  = threadIdx.x;
    const int lane = tid & (WAVE - 1);
    const int wid  = tid >> 5;
    const int nw   = blockDim.x >> 5;

    // --- max ---
    float m = -INFINITY;
    for (int i = tid; i < N; i += blockDim.x) m = fmaxf(m, logits[i]);
    #pragma unroll
    for (int off = WAVE / 2; off > 0; off >>= 1)
        m = fmaxf(m, __shfl_xor(m, off, WAVE));
    if (lane == 0) red[wid] = m;
    __syncthreads();
    if (tid == 0) {
        float mm = -INFINITY;
        for (int i = 0; i < nw; ++i) mm = fmaxf(mm, red[i]);
        bcast = mm;
    }
    __syncthreads();
    const float gmax = bcast;
    __syncthreads();

    // --- sum of exp ---
    float s = 0.0f;
    for (int i = tid; i < N; i += blockDim.x) s += __expf(logits[i] - gmax);
    #pragma unroll
    for (int off = WAVE / 2; off > 0; off >>= 1)
        s += __shfl_xor(s, off, WAVE);
    if (lane == 0) red[wid] = s;
    __syncthreads();
    if (tid == 0) {
        float ss = 0.0f;
        for (int i = 0; i < nw; ++i) ss += red[i];
        bcast = ss;
    }
    __syncthreads();
    const float inv = 1.0f / bcast;

    for (int i = tid; i < N; i += blockDim.x)
        out[i] = __expf(logits[i] - gmax) * inv;
}

// ---------------------------------------------------------------------------
extern "C" void kernel_launch(void* const* d_in, const int* in_sizes, int n_in,
                              void* d_out, int out_size, void* d_ws, size_t ws_size,
                              hipStream_t stream) {
    // dims
    constexpr int V = 10000, E = 1000, H = 1000, D = 2048, A = 512, R = 36;
    constexpr int KATT = H + D + E;   // 4048
    constexpr int KLANG = D + H;      // 3048

    const float* onehot    = (const float*)d_in[0];
    const float* hlangprev = (const float*)d_in[1];
    const float* pooled    = (const float*)d_in[2];
    const float* img       = (const float*)d_in[3];   // (1,36,2048)
    const float* W_emb     = (const float*)d_in[4];   // (E,V)
    const float* b_emb     = (const float*)d_in[5];
    const float* W_ih_att  = (const float*)d_in[6];   // (4H, 4048)
    /* d_in[7] W_hh_att unused (h0 = 0) */
    const float* b_ih_att  = (const float*)d_in[8];
    const float* b_hh_att  = (const float*)d_in[9];
    const float* Wf        = (const float*)d_in[10];  // (A, D)
    const float* Wh        = (const float*)d_in[11];  // (A, H)
    const float* Wa        = (const float*)d_in[12];  // (1, A)
    const float* W_ih_lang = (const float*)d_in[13];  // (4H, 3048)
    /* d_in[14] W_hh_lang unused */
    const float* b_ih_lang = (const float*)d_in[15];
    const float* b_hh_lang = (const float*)d_in[16];
    const float* W_sel     = (const float*)d_in[17];  // (V, H)
    const float* b_sel     = (const float*)d_in[18];
    float* out = (float*)d_out;

    // workspace layout (floats)
    float* ws = (float*)d_ws;
    float* x_att      = ws;               // 4048
    float* gates_att  = x_att + KATT;     // 4000
    float* h_att      = gates_att + 4*H;  // 1000
    float* wf         = h_att + H;        // 36*512 = 18432
    float* wh         = wf + R * A;       // 512
    float* aw         = wh + A;           // 36
    float* x_lang     = aw + R;           // 3048
    float* gates_lang = x_lang + KLANG;   // 4000
    float* h_lang     = gates_lang + 4*H; // 1000
    float* logits     = h_lang + H;       // 10000

    // 1) x_att[0:3048) = concat(h_lang_prev, pooled)
    concat_xatt_kernel<<<(3048 + 255) / 256, 256, 0, stream>>>(hlangprev, pooled, x_att);
    // 2) prev_word = W_emb @ onehot + b_emb  ->  x_att[3048:4048)
    gemv_kernel<<<E, 256, 0, stream>>>(W_emb, onehot, b_emb, nullptr, x_att + H + D, V);
    // 3) attention LSTM gates = x_att @ W_ih_att^T + b_ih + b_hh
    gemv_kernel<<<4 * H, 256, 0, stream>>>(W_ih_att, x_att, b_ih_att, b_hh_att, gates_att, KATT);
    // 4) h_att
    lstm_act_kernel<<<(H + 255) / 256, 256, 0, stream>>>(gates_att, h_att, H);
    // 5) wf = img @ Wf^T  (WMMA f32 16x16x4; 32 N-tiles x 3 M-tiles)
    wf_wmma_kernel<<<dim3(A / 16, (R + 15) / 16), 32, 0, stream>>>(img, Wf, wf);
    // 6) wh = Wh @ h_att
    gemv_kernel<<<A, 256, 0, stream>>>(Wh, h_att, nullptr, nullptr, wh, H);
    // 7) aw[r] = Wa . tanh(wf[r] + wh)
    att_score_kernel<<<R, 256, 0, stream>>>(wf, wh, Wa, aw);
    // 8) x_lang = concat(sum_r aw[r]*img[r], h_att)
    attend_kernel<<<(KLANG + 255) / 256, 256, 0, stream>>>(img, aw, h_att, x_lang);
    // 9) language LSTM gates
    gemv_kernel<<<4 * H, 256, 0, stream>>>(W_ih_lang, x_lang, b_ih_lang, b_hh_lang, gates_lang, KLANG);
    // 10) h_lang
    lstm_act_kernel<<<(H + 255) / 256, 256, 0, stream>>>(gates_lang, h_lang, H);
    // 11) logits = W_sel @ h_lang + b_sel
    gemv_kernel<<<V, 256, 0, stream>>>(W_sel, h_lang, b_sel, nullptr, logits, H);
    // 12) softmax -> output probabilities
    softmax_kernel<<<1, 1024, 0, stream>>>(logits, out, V);

    (void)in_sizes; (void)n_in; (void)out_size; (void)ws_size;
}